// GeoSparseLinear_5377299054825
// MI455X (gfx1250) — compile-verified
//
#include <hip/hip_runtime.h>

// GeoSparseLinear: out[b,o] = sum_k x[b, idx[o,k]] * w[o,k] + bias[o]
// B=4096, IN_F=8192, OUT_F=4096, K=32.  fp32 everywhere.
//
// Roofline: ~193 MB compulsory HBM traffic (x read once: 128 MB, out: 64 MB)
// => ~8us floor @ 23.3 TB/s.  The hard part is 536M random 4B gathers; we
// serve them from LDS.  Per block: stage ROWS=4 batch rows of x into LDS as
// 16B-aligned column quads lds[c*4+r] via the CDNA5 async global->LDS engine
// (ASYNCcnt), then each thread owns one output o and one ds_load_b128 fetches
// the 4 batch values of column idx[o,k].  x is read from HBM exactly once.
//
// WMMA deliberately not used: the contraction is block-diagonal (each output
// touches only its own 32 gathered features), so dense WMMA runs at 1/16
// density -- exact-f32 V_WMMA_F32_16X16X4_F32 at 16x waste would turn this
// memory-bound kernel compute-bound, and the fp32 transposed mapping needs
// 16 staged rows = 512 KB > 320 KB LDS.

#define IN_F   8192
#define OUT_F  4096
#define BATCH  4096
#define KS     32
#define ROWS   4
#define STRIDE 4          // dwords per column group -> 16B aligned for b128
#define THREADS 256
#define STAGE_ITERS ((ROWS * IN_F) / THREADS)   // 128

typedef float v4f __attribute__((ext_vector_type(4)));
typedef __attribute__((address_space(1))) int as1_int;
typedef __attribute__((address_space(3))) int as3_int;

__global__ __launch_bounds__(THREADS, 2)
void gsl_kernel(const float* __restrict__ x,
                const float* __restrict__ weight,
                const float* __restrict__ bias,
                const int*   __restrict__ idx,
                float*       __restrict__ out)
{
    __shared__ float lds[IN_F * STRIDE];   // 128 KB -> 2 blocks/WGP

    const int tid  = threadIdx.x;
    const int row0 = blockIdx.x * ROWS;
    const float* xrow = x + (size_t)row0 * IN_F;   // 4 consecutive rows

    // ---- Stage 4 rows of x into LDS, transposed into column quads:
    // flat = r*IN_F + c walks global memory contiguously (lane-coalesced);
    // LDS side scatters to lds[c*4 + r].  Fully unrolled: straight-line async
    // loads, r and all offsets are compile-time constants per step.
#if __has_builtin(__builtin_amdgcn_global_load_async_to_lds_b32) && \
    __has_builtin(__builtin_amdgcn_s_wait_asynccnt)
    #pragma unroll
    for (int i = 0; i < STAGE_ITERS; ++i) {
        const int flat = i * THREADS + tid;        // 0..ROWS*IN_F-1
        const int c    = flat & (IN_F - 1);
        const int r    = flat >> 13;               // compile-time per i
        as1_int* g = (as1_int*)(xrow + flat);
        as3_int* l = (as3_int*)(&lds[c * STRIDE + r]);
        __builtin_amdgcn_global_load_async_to_lds_b32(g, l, 0, 0);
    }
    __builtin_amdgcn_s_wait_asynccnt(0);
#else
    #pragma unroll 8
    for (int i = 0; i < STAGE_ITERS; ++i) {
        const int flat = i * THREADS + tid;
        const int c    = flat & (IN_F - 1);
        const int r    = flat >> 13;
        lds[c * STRIDE + r] = xrow[flat];
    }
#endif
    __syncthreads();

    // ---- Each thread computes 16 outputs (o strided by 256) for the 4 rows.
    for (int ob = 0; ob < OUT_F / THREADS; ++ob) {
        const int o = ob * THREADS + tid;
        const int*   ip = idx    + (size_t)o * KS;   // 128 B per thread, L2-hot
        const float* wp = weight + (size_t)o * KS;
        // Warm the NEXT ob iteration's rows (+THREADS outputs = +32 KB).
        // Speculative prefetch: silently dropped if past end of buffer.
        __builtin_prefetch(ip + (size_t)THREADS * KS, 0, 0);  // global_prefetch_b8
        __builtin_prefetch(wp + (size_t)THREADS * KS, 0, 0);

        float a0 = 0.f, a1 = 0.f, a2 = 0.f, a3 = 0.f;
        #pragma unroll
        for (int k = 0; k < KS; ++k) {
            const int  c  = ip[k];                           // 0..8191
            const v4f  xv = *(const v4f*)(&lds[c * STRIDE]); // ds_load_b128
            const float wk = wp[k];
            a0 = fmaf(xv.x, wk, a0);
            a1 = fmaf(xv.y, wk, a1);
            a2 = fmaf(xv.z, wk, a2);
            a3 = fmaf(xv.w, wk, a3);
        }
        const float bo = bias[o];
        float* op = out + (size_t)row0 * OUT_F + o;   // lanes contiguous in o
        op[(size_t)0 * OUT_F] = a0 + bo;
        op[(size_t)1 * OUT_F] = a1 + bo;
        op[(size_t)2 * OUT_F] = a2 + bo;
        op[(size_t)3 * OUT_F] = a3 + bo;
    }
}

extern "C" void kernel_launch(void* const* d_in, const int* in_sizes, int n_in,
                              void* d_out, int out_size, void* d_ws, size_t ws_size,
                              hipStream_t stream) {
    (void)in_sizes; (void)n_in; (void)out_size; (void)d_ws; (void)ws_size;
    const float* x      = (const float*)d_in[0];
    const float* weight = (const float*)d_in[1];
    const float* bias   = (const float*)d_in[2];
    const int*   idx    = (const int*)d_in[3];
    float* out = (float*)d_out;

    dim3 grid(BATCH / ROWS);   // 1024 blocks
    gsl_kernel<<<grid, THREADS, 0, stream>>>(x, weight, bias, idx, out);
}